// SubjectLayer_75608604279341
// MI455X (gfx1250) — compile-verified
//
#include <hip/hip_runtime.h>

typedef __attribute__((ext_vector_type(2))) float v2f;
typedef __attribute__((ext_vector_type(8))) float v8f;

// out[b,o,t] = sum_c W[s(b),o,c] * x[b,c,t]
// B=128, C=272 (= 17*16), T=1024 (= 64*16), fp32, V_WMMA_F32_16X16X4_F32.
//
// One wave -> 16(M) x 64(N) output slab for one batch sample.
// Wave id decomposition: [b][m-tile 0..16][n-group 0..15], 8 waves / block.

#define C_DIM 272
#define T_DIM 1024
#define M_TILES 17
#define N_GROUPS 16   // 16 groups x 64 cols = 1024
#define WAVES_PER_BLOCK 8

__global__ __launch_bounds__(256)
void subject_mix_wmma_f32(const float* __restrict__ x,
                          const int* __restrict__ subject_idx,
                          const float* __restrict__ W,
                          float* __restrict__ out) {
    const int lane = threadIdx.x & 31;
    const int wave = threadIdx.x >> 5;
    const int wid  = blockIdx.x * WAVES_PER_BLOCK + wave;

    const int ngrp = wid & (N_GROUPS - 1);          // 0..15
    const int m    = (wid >> 4) % M_TILES;          // 0..16
    const int b    = wid / (N_GROUPS * M_TILES);    // 0..127

    const int s    = subject_idx[b];
    const int r    = lane & 15;                     // row/col within tile
    const int half = lane >> 4;                     // half-wave K split

    // A operand: W[s, m*16 + r, k0 + 2*half + {0,1}]  (contiguous pair -> b64)
    const float* __restrict__ Wrow =
        W + (size_t)s * (C_DIM * C_DIM) + (size_t)(m * 16 + r) * C_DIM + 2 * half;

    // B operand: x[b, k0 + 2*half + {0,1}, nbase + q*16 + r]
    const int nbase = ngrp * 64;
    const float* __restrict__ xb =
        x + (size_t)b * (C_DIM * T_DIM) + (size_t)(2 * half) * T_DIM + nbase + r;

    v8f acc0 = {}, acc1 = {}, acc2 = {}, acc3 = {};

    #pragma unroll 4
    for (int k0 = 0; k0 < C_DIM; k0 += 4) {
        v2f a = *(const v2f*)(Wrow + k0);           // global_load_b64

        const float* __restrict__ xk = xb + (size_t)k0 * T_DIM;
        v2f b0, b1, b2, b3;
        b0.x = xk[0];            b0.y = xk[T_DIM];
        b1.x = xk[16];           b1.y = xk[T_DIM + 16];
        b2.x = xk[32];           b2.y = xk[T_DIM + 32];
        b3.x = xk[48];           b3.y = xk[T_DIM + 48];

        acc0 = __builtin_amdgcn_wmma_f32_16x16x4_f32(
            false, a, false, b0, (short)0, acc0, false, false);
        acc1 = __builtin_amdgcn_wmma_f32_16x16x4_f32(
            false, a, false, b1, (short)0, acc1, false, false);
        acc2 = __builtin_amdgcn_wmma_f32_16x16x4_f32(
            false, a, false, b2, (short)0, acc2, false, false);
        acc3 = __builtin_amdgcn_wmma_f32_16x16x4_f32(
            false, a, false, b3, (short)0, acc3, false, false);
    }

    // D layout: VGPR i -> row m*16 + i + 8*half, col = nbase + q*16 + r
    float* __restrict__ ob =
        out + (size_t)b * (C_DIM * T_DIM) + (size_t)(m * 16 + 8 * half) * T_DIM + nbase + r;

    #pragma unroll
    for (int i = 0; i < 8; ++i) {
        ob[(size_t)i * T_DIM +  0] = acc0[i];
        ob[(size_t)i * T_DIM + 16] = acc1[i];
        ob[(size_t)i * T_DIM + 32] = acc2[i];
        ob[(size_t)i * T_DIM + 48] = acc3[i];
    }
}

extern "C" void kernel_launch(void* const* d_in, const int* in_sizes, int n_in,
                              void* d_out, int out_size, void* d_ws, size_t ws_size,
                              hipStream_t stream) {
    const float* x   = (const float*)d_in[0];   // [128, 272, 1024]
    const int*   idx = (const int*)d_in[1];     // [128]
    const float* W   = (const float*)d_in[2];   // [16, 272, 272]
    float*       out = (float*)d_out;           // [128, 272, 1024]

    (void)in_sizes; (void)n_in; (void)out_size; (void)d_ws; (void)ws_size;

    const int total_waves = 128 * M_TILES * N_GROUPS;        // 34816
    const int blocks      = total_waves / WAVES_PER_BLOCK;   // 4352

    subject_mix_wmma_f32<<<blocks, 256, 0, stream>>>(x, idx, W, out);
}